// ParallelNoise_42958262894942
// MI455X (gfx1250) — compile-verified
//
#include <hip/hip_runtime.h>
#include <stdint.h>

// Problem geometry (from reference)
#define B_IMG  64
#define C_CH   3
#define HDIM   224
#define HW     (HDIM*HDIM)        // 50176 = 196*256 (exactly divisible -> no EXEC divergence)
#define DFEAT  (C_CH*HW)          // 150528 = 588*256
#define NTOT   (B_IMG*DFEAT)      // 9633792
#define HALF_N ((uint32_t)(NTOT/2)) // 4816896 : threefry pairs element i with i+HALF_N (= image b with b+32)
#define NUM_K  64
#define SIGMA_C 0.05f
#define EPS_C   1e-10f

typedef float v2f __attribute__((ext_vector_type(2)));
typedef float v8f __attribute__((ext_vector_type(8)));

// ---------------- JAX-exact threefry2x32 (20 rounds) ----------------
__device__ __forceinline__ uint32_t rotl32(uint32_t x, int r){ return (x<<r)|(x>>(32-r)); }

__device__ __forceinline__ void threefry2x32(uint32_t k0, uint32_t k1, uint32_t &x0, uint32_t &x1){
  const uint32_t k2 = k0 ^ k1 ^ 0x1BD11BDAu;
  x0 += k0; x1 += k1;
#define TF_R(r) { x0 += x1; x1 = rotl32(x1,(r)); x1 ^= x0; }
  TF_R(13) TF_R(15) TF_R(26) TF_R(6)  x0 += k1; x1 += k2 + 1u;
  TF_R(17) TF_R(29) TF_R(16) TF_R(24) x0 += k2; x1 += k0 + 2u;
  TF_R(13) TF_R(15) TF_R(26) TF_R(6)  x0 += k0; x1 += k1 + 3u;
  TF_R(17) TF_R(29) TF_R(16) TF_R(24) x0 += k1; x1 += k2 + 4u;
  TF_R(13) TF_R(15) TF_R(26) TF_R(6)  x0 += k2; x1 += k0 + 5u;
#undef TF_R
}

// bits -> uniform[-0.99999994, 1) -> sqrt(2)*erfinv(u)  (JAX _normal_real)
__device__ __forceinline__ float bits_to_normal(uint32_t b){
  const float lo = -0.99999994f;                       // nextafter(-1,0); span rounds to 2.0f
  float f = __uint_as_float((b >> 9) | 0x3f800000u) - 1.0f;  // [0,1)
  float u = fmaxf(fmaf(f, 2.0f, lo), lo);
  return 1.4142135f * erfinvf(u);
}

// channel-normalize a triple: (t-mu)/sqrt(var_ddof1) * SIGMA
__device__ __forceinline__ void norm3(float &a, float &b, float &c){
  float mu = (a + b + c) * (1.0f/3.0f);
  float da = a - mu, db = b - mu, dc = c - mu;
  float var = (da*da + db*db + dc*dc) * 0.5f;          // n-1 = 2
  float s = SIGMA_C * rsqrtf(var);
  a = da * s; b = db * s; c = dc * s;
}

__device__ __forceinline__ float sigmoidf(float x){ return 1.0f / (1.0f + __expf(-x)); }

// -------- wave32 reduction on the matrix pipe: V_WMMA_F32_16X16X4_F32 --------
// A VGPR0: lanes0-15 -> (M=lane,K=0), lanes16-31 -> (M=lane-16,K=2); A VGPR1 zeroed.
// B = ones  =>  D[m,n] = v_m + v_{m+16}.  Sum the 8 C regs per lane, then xor-16 shuffle.
__device__ __forceinline__ float wave_reduce(float v){
  v2f a; a.x = v;    a.y = 0.0f;
  v2f bo; bo.x = 1.0f; bo.y = 1.0f;
  v8f c = {};
  c = __builtin_amdgcn_wmma_f32_16x16x4_f32(false, a, false, bo, (short)0, c, false, false);
  float s = c[0]+c[1]+c[2]+c[3]+c[4]+c[5]+c[6]+c[7];
  s += __shfl_xor(s, 16, 32);
  return s;  // full-wave sum, uniform across lanes
}

// ---------------- K0: 64 split keys of key(42) (exact JAX split) ----------------
__global__ void k0_keys(uint32_t* __restrict__ keys){
  int t = threadIdx.x;              // 0..63
  if (t < 32){
    uint32_t a0 = 2u*(uint32_t)t,      a1 = a0 + 64u;
    uint32_t b0 = 2u*(uint32_t)t + 1u, b1 = b0 + 64u;
    threefry2x32(0u, 42u, a0, a1);
    threefry2x32(0u, 42u, b0, b1);
    keys[2*t] = a0; keys[2*t+1] = b0;           // from x0 half
  } else {
    uint32_t j  = 2u*(uint32_t)t - 64u;
    uint32_t a0 = j,      a1 = j + 64u;
    uint32_t b0 = j + 1u, b1 = b0 + 64u;
    threefry2x32(0u, 42u, a0, a1);
    threefry2x32(0u, 42u, b0, b1);
    keys[2*t] = a1; keys[2*t+1] = b1;           // from x1 half
  }
}

// ---------------- K1: P0[b] = <clip(img+noise), w> ----------------
__global__ void k1_base_dot(const float* __restrict__ img, const float* __restrict__ noi,
                            const float* __restrict__ w, float* __restrict__ P0){
  int b = blockIdx.x, tid = threadIdx.x;
  const float* ib = img + (size_t)b * DFEAT;
  const float* nb = noi + (size_t)b * DFEAT;
  float acc = 0.f;
  for (int i = tid; i < DFEAT; i += 256){        // 588 exact iterations
    float v = fminf(fmaxf(ib[i] + nb[i], 0.f), 1.f);
    acc = fmaf(v, w[i], acc);
  }
  float r = wave_reduce(acc);
  __shared__ float red[8];
  if ((tid & 31) == 0) red[tid >> 5] = r;
  __syncthreads();
  if (tid == 0){
    float s = 0.f;
    #pragma unroll
    for (int i = 0; i < 8; ++i) s += red[i];
    P0[b] = s;
  }
}

// ---------------- K2: S[k][b] = <clip(img+noise+t_k), w>  (b and b+32 per block) ----------------
__global__ void k2_probe_dot(const float* __restrict__ img, const float* __restrict__ noi,
                             const float* __restrict__ w, const uint32_t* __restrict__ keys,
                             float* __restrict__ S){
  const int bp  = blockIdx.x;        // 0..31
  const int k   = blockIdx.y;        // 0..63
  const int tid = threadIdx.x;
  const uint32_t k0 = keys[2*k], k1 = keys[2*k+1];
  const uint32_t basec = (uint32_t)bp * (uint32_t)DFEAT;
  float acc0 = 0.f, acc1 = 0.f;
  for (int p = tid; p < HW; p += 256){           // 196 exact iterations
    float n0[3], n1[3];
    #pragma unroll
    for (int c = 0; c < 3; ++c){
      uint32_t x0 = basec + (uint32_t)(c*HW) + (uint32_t)p;
      uint32_t x1 = x0 + HALF_N;
      threefry2x32(k0, k1, x0, x1);
      n0[c] = bits_to_normal(x0);                // element for image bp
      n1[c] = bits_to_normal(x1);                // element for image bp+32
    }
    norm3(n0[0], n0[1], n0[2]);
    norm3(n1[0], n1[1], n1[2]);
    #pragma unroll
    for (int c = 0; c < 3; ++c){
      size_t i0 = (size_t)bp*DFEAT + (size_t)(c*HW) + (size_t)p;
      size_t i1 = i0 + (size_t)HALF_N;
      float wv = w[c*HW + p];
      float v0 = fminf(fmaxf(img[i0] + noi[i0] + n0[c], 0.f), 1.f);
      float v1 = fminf(fmaxf(img[i1] + noi[i1] + n1[c], 0.f), 1.f);
      acc0 = fmaf(v0, wv, acc0);
      acc1 = fmaf(v1, wv, acc1);
    }
  }
  float r0 = wave_reduce(acc0);
  float r1 = wave_reduce(acc1);
  __shared__ float red[16];
  if ((tid & 31) == 0){ red[tid >> 5] = r0; red[8 + (tid >> 5)] = r1; }
  __syncthreads();
  if (tid == 0){
    float s0 = 0.f, s1 = 0.f;
    #pragma unroll
    for (int i = 0; i < 8; ++i){ s0 += red[i]; s1 += red[8+i]; }
    S[k*64 + bp]      = s0;
    S[k*64 + bp + 32] = s1;
  }
}

// ---------------- K3: d[b][k], and scale[b] = 1/(64*(sqrt(nc)+eps)*sigma) ----------------
__global__ void k3_dvec(const float* __restrict__ P0, const float* __restrict__ S,
                        const float* __restrict__ bptr,
                        float* __restrict__ dmat, float* __restrict__ scale){
  int b = blockIdx.x, k = threadIdx.x;           // 64 x 64
  float b0 = bptr[0];
  float prob2 = sigmoidf(P0[b] + b0);
  float dv = prob2 - sigmoidf(S[k*64 + b] + b0);
  dmat[b*64 + k] = dv;
  __shared__ float sh[64];
  sh[k] = dv * dv;
  __syncthreads();
  for (int off = 32; off > 0; off >>= 1){
    if (k < off) sh[k] += sh[k + off];
    __syncthreads();
  }
  if (k == 0){
    float nc = sh[0];
    scale[b] = 1.0f / ((float)NUM_K * (sqrtf(nc) + EPS_C) * SIGMA_C);
  }
}

// ---------------- K4: out = (sum_k t_k * d_k) * scale ----------------
__global__ void k4_output(const uint32_t* __restrict__ keys, const float* __restrict__ dmat,
                          const float* __restrict__ scale, float* __restrict__ out){
  const int bp  = blockIdx.y;                    // 0..31
  const int tid = threadIdx.x;
  const int p   = blockIdx.x * 256 + tid;        // HW = 196*256, no tail
  __shared__ float    d0s[64], d1s[64];
  __shared__ uint32_t ks[128];
  if (tid < 64){ d0s[tid] = dmat[bp*64 + tid]; d1s[tid] = dmat[(bp+32)*64 + tid]; }
  if (tid >= 64 && tid < 192) ks[tid - 64] = keys[tid - 64];
  __syncthreads();
  const uint32_t basec = (uint32_t)bp * (uint32_t)DFEAT;
  float wd0[3] = {0.f,0.f,0.f}, wd1[3] = {0.f,0.f,0.f};
  for (int k = 0; k < NUM_K; ++k){
    uint32_t kk0 = ks[2*k], kk1 = ks[2*k+1];
    float n0[3], n1[3];
    #pragma unroll
    for (int c = 0; c < 3; ++c){
      uint32_t x0 = basec + (uint32_t)(c*HW) + (uint32_t)p;
      uint32_t x1 = x0 + HALF_N;
      threefry2x32(kk0, kk1, x0, x1);
      n0[c] = bits_to_normal(x0);
      n1[c] = bits_to_normal(x1);
    }
    norm3(n0[0], n0[1], n0[2]);
    norm3(n1[0], n1[1], n1[2]);
    float dv0 = d0s[k], dv1 = d1s[k];
    #pragma unroll
    for (int c = 0; c < 3; ++c){
      wd0[c] = fmaf(n0[c], dv0, wd0[c]);
      wd1[c] = fmaf(n1[c], dv1, wd1[c]);
    }
  }
  float s0 = scale[bp], s1 = scale[bp+32];
  #pragma unroll
  for (int c = 0; c < 3; ++c){
    size_t i0 = (size_t)bp*DFEAT + (size_t)(c*HW) + (size_t)p;
    out[i0]                   = wd0[c] * s0;
    out[i0 + (size_t)HALF_N]  = wd1[c] * s1;
  }
}

extern "C" void kernel_launch(void* const* d_in, const int* in_sizes, int n_in,
                              void* d_out, int out_size, void* d_ws, size_t ws_size,
                              hipStream_t stream){
  (void)in_sizes; (void)n_in; (void)out_size; (void)ws_size;
  const float* img = (const float*)d_in[0];   // [64,3,224,224]
  const float* noi = (const float*)d_in[1];   // [64,3,224,224]
  const float* w   = (const float*)d_in[2];   // [150528]
  const float* b0  = (const float*)d_in[3];   // [1]
  float* out = (float*)d_out;

  // workspace layout (bytes): keys 512 | P0 256 | S 16K | dmat 16K | scale 256
  uint32_t* keys  = (uint32_t*)d_ws;
  float*    wsf   = (float*)d_ws;
  float*    P0    = wsf + 128;
  float*    S     = wsf + 192;
  float*    dmat  = wsf + 192 + 64*64;
  float*    scale = wsf + 192 + 2*64*64;

  hipLaunchKernelGGL(k0_keys,      dim3(1),       dim3(64),  0, stream, keys);
  hipLaunchKernelGGL(k1_base_dot,  dim3(64),      dim3(256), 0, stream, img, noi, w, P0);
  hipLaunchKernelGGL(k2_probe_dot, dim3(32, 64),  dim3(256), 0, stream, img, noi, w, keys, S);
  hipLaunchKernelGGL(k3_dvec,      dim3(64),      dim3(64),  0, stream, P0, S, b0, dmat, scale);
  hipLaunchKernelGGL(k4_output,    dim3(196, 32), dim3(256), 0, stream, keys, dmat, scale, out);
}